// moratt_38371237822848
// MI455X (gfx1250) — compile-verified
//
#include <hip/hip_runtime.h>
#include <math.h>

// MorAtt fused pipeline for MI455X (gfx1250, wave32).
// B=8, C=64, H=W=256. fp32 throughout (matches reference).
// GEMM passes use V_WMMA_F32_16X16X4_F32 (fp32 matrix cores).
// Tile staging uses GLOBAL_LOAD_ASYNC_TO_LDS_B128 (ASYNCcnt-tracked direct
// global->LDS path) when the builtin exists, else b128 via VGPRs.

typedef __attribute__((ext_vector_type(2))) float v2f;
typedef __attribute__((ext_vector_type(8))) float v8f;

#define NB   8
#define NC   64
#define NH   256
#define NW   256
#define NHW  65536      // 256*256
#define SXS  132        // LDS x-tile stride (128 -> 132; 528B rows, 16B aligned)
#define SWS  68         // LDS weight stride (64 -> 68; 272B rows, 16B aligned)
#define BIG  1.0e4f

__device__ __forceinline__ float silu_f(float g) { return g / (1.0f + __expf(-g)); }

#if __has_builtin(__builtin_amdgcn_global_load_async_to_lds_b128)
#define HAVE_ASYNC_LDS 1
// Match the builtin's parameter types exactly:
//   (v4i __attribute__((address_space(1)))*, v4i __attribute__((address_space(3)))*, imm, imm)
typedef int v4i_gen __attribute__((vector_size(16)));       // generic vector (not ext_vector)
typedef __attribute__((address_space(1))) v4i_gen* g128_p;  // global (printed as __device__)
typedef __attribute__((address_space(3))) v4i_gen* l128_p;  // LDS (shared)
__device__ __forceinline__ void async_copy16(const float* g, float* l) {
    __builtin_amdgcn_global_load_async_to_lds_b128((g128_p)g, (l128_p)l, 0, 0);
}
__device__ __forceinline__ void async_wait() {
#if __has_builtin(__builtin_amdgcn_s_wait_asynccnt)
    __builtin_amdgcn_s_wait_asynccnt(0);
#else
    asm volatile("s_wait_asynccnt 0x0" ::: "memory");
#endif
}
#else
#define HAVE_ASYNC_LDS 0
#endif

// ---------------------------------------------------------------------------
// K0: zero A2 accumulator (graph-safe deterministic init)
// ---------------------------------------------------------------------------
__global__ void k_zero(float* p, int n) {
    int i = blockIdx.x * blockDim.x + threadIdx.x;
    if (i < n) p[i] = 0.0f;
}

// ---------------------------------------------------------------------------
// K1: per-(b,c) stats on Haar A-band: mean(A) and max(A), float4 loads.
// A[i,j] = 0.5*(x[2i,2j]+x[2i+1,2j]+x[2i,2j+1]+x[2i+1,2j+1])
// ---------------------------------------------------------------------------
__global__ __launch_bounds__(256) void k_stats(const float* __restrict__ x,
                                               float* __restrict__ gapraw,
                                               float* __restrict__ gmpraw) {
    int bc = blockIdx.x;                 // 0..511
    const float* p = x + ((size_t)bc << 16);
    int tid = threadIdx.x;
    float sum = 0.0f, mx = -3.4e38f;
    // 8192 float4-pairs: each j covers two adjacent A elements (4x2 x-patch)
    for (int j = tid; j < 8192; j += 256) {
        int ay = j >> 6, ax4 = j & 63;
        const float4 r0 = *(const float4*)(p + (ay * 2) * NW + ax4 * 4);
        const float4 r1 = *(const float4*)(p + (ay * 2 + 1) * NW + ax4 * 4);
        float s4a = r0.x + r0.y + r1.x + r1.y;
        float s4b = r0.z + r0.w + r1.z + r1.w;
        sum += s4a + s4b;
        mx = fmaxf(mx, 0.5f * fmaxf(s4a, s4b));
    }
    __shared__ float rs[256], rm[256];
    rs[tid] = sum; rm[tid] = mx;
    __syncthreads();
    for (int s = 128; s > 0; s >>= 1) {
        if (tid < s) { rs[tid] += rs[tid + s]; rm[tid] = fmaxf(rm[tid], rm[tid + s]); }
        __syncthreads();
    }
    if (tid == 0) {
        gapraw[bc] = 0.5f * rs[0] / 16384.0f;   // mean(A)
        gmpraw[bc] = rm[0];                     // max(A)
    }
}

// ---------------------------------------------------------------------------
// K2: softmax over channels (64) for gap and gmp. 1 block per b, 64 threads.
// ---------------------------------------------------------------------------
__global__ void k_softmax_c(const float* __restrict__ gapraw,
                            const float* __restrict__ gmpraw,
                            float* __restrict__ gap, float* __restrict__ gmp) {
    int b = blockIdx.x, tid = threadIdx.x;
    __shared__ float red[64];
    float ga = gapraw[b * NC + tid];
    float gm = gmpraw[b * NC + tid];
    // softmax(ga)
    red[tid] = ga; __syncthreads();
    for (int s = 32; s > 0; s >>= 1) { if (tid < s) red[tid] = fmaxf(red[tid], red[tid + s]); __syncthreads(); }
    float mx = red[0]; __syncthreads();
    float e = __expf(ga - mx);
    red[tid] = e; __syncthreads();
    for (int s = 32; s > 0; s >>= 1) { if (tid < s) red[tid] += red[tid + s]; __syncthreads(); }
    gap[b * NC + tid] = e / red[0];
    __syncthreads();
    // softmax(gm)
    red[tid] = gm; __syncthreads();
    for (int s = 32; s > 0; s >>= 1) { if (tid < s) red[tid] = fmaxf(red[tid], red[tid + s]); __syncthreads(); }
    mx = red[0]; __syncthreads();
    e = __expf(gm - mx);
    red[tid] = e; __syncthreads();
    for (int s = 32; s > 0; s >>= 1) { if (tid < s) red[tid] += red[tid + s]; __syncthreads(); }
    gmp[b * NC + tid] = e / red[0];
}

// ---------------------------------------------------------------------------
// Shared staging: W[64x64] and x-tile[64x128] -> LDS (async if available)
// ---------------------------------------------------------------------------
__device__ __forceinline__ void stage_tiles(const float* __restrict__ convw,
                                            const float* __restrict__ xb,
                                            int base, float* sW, float* sX,
                                            int tid) {
#if HAVE_ASYNC_LDS
    for (int i = tid; i < 1024; i += 256) {          // 4096 floats of W
        int o = i >> 4, c4 = i & 15;
        async_copy16(convw + i * 4, &sW[o * SWS + c4 * 4]);
    }
    for (int i = tid; i < 2048; i += 256) {          // 64x128 x-tile
        int c = i >> 5, n4 = i & 31;
        async_copy16(xb + (size_t)c * NHW + base + n4 * 4, &sX[c * SXS + n4 * 4]);
    }
    async_wait();
#else
    for (int i = tid; i < 1024; i += 256) {
        int o = i >> 4, c4 = i & 15;
        *(float4*)&sW[o * SWS + c4 * 4] = ((const float4*)convw)[i];
    }
    for (int i = tid; i < 2048; i += 256) {
        int c = i >> 5, n4 = i & 31;
        *(float4*)&sX[c * SXS + n4 * 4] =
            *(const float4*)(xb + (size_t)c * NHW + base + n4 * 4);
    }
#endif
}

// ---------------------------------------------------------------------------
// K3: GEMM pass 1 (WMMA). Per 128-pixel chunk:
//   v = silu(conv_w @ x * s + b) [never stored]
//   vgap[n] = sum_c gap[c] v[c,n] ; vgmp[n] = sum_c gmp[c] v[c,n]
//   xs[n]   = silu((convA_w . x[:,n]) * sA + bA)
// Block = 256 threads = 8 waves; wave w owns pixels [16w,16w+16).
// WMMA f32 16x16x4, K chained 16x (K=64), 4 output tiles of 16 channels.
// ---------------------------------------------------------------------------
__global__ __launch_bounds__(256) void k_gemm_pass1(
    const float* __restrict__ x,
    const float* __restrict__ convw, const float* __restrict__ convs,
    const float* __restrict__ convb,
    const float* __restrict__ convAw, const float* __restrict__ convAs,
    const float* __restrict__ convAb,
    const float* __restrict__ gap, const float* __restrict__ gmp,
    float* __restrict__ vgap, float* __restrict__ vgmp, float* __restrict__ xs) {
    __shared__ float sW[NC * SWS];
    __shared__ float sX[NC * SXS];
    __shared__ float sS[NC], sB[NC], sG[NC], sM[NC], sCA[NC];

    int tid = threadIdx.x;
    int b = blockIdx.x >> 9;            // 512 chunks per batch
    int chunk = blockIdx.x & 511;
    int base = chunk * 128;
    const float* xb = x + ((size_t)b * NC) * NHW;

    if (tid < NC) {
        sS[tid] = convs[tid]; sB[tid] = convb[tid];
        sG[tid] = gap[b * NC + tid]; sM[tid] = gmp[b * NC + tid];
        sCA[tid] = convAw[tid];
    }
    stage_tiles(convw, xb, base, sW, sX, tid);
    __syncthreads();

    int lane = tid & 31;
    int wv = tid >> 5;
    int n0 = wv * 16;
    int mrow = lane & 15;              // A row (M) == B col (N)
    int khalf = (lane >> 4) << 1;      // 0 (lanes 0-15) or 2 (lanes 16-31)
    float vg = 0.0f, vm = 0.0f;

    for (int t = 0; t < 4; ++t) {
        int o0 = t * 16;
        v8f acc = {};
        #pragma unroll
        for (int s = 0; s < 16; ++s) {
            v2f a, bb;
            // A: 16x4 fragment of conv_w[o0+m][4s+k]
            a.x = sW[(o0 + mrow) * SWS + 4 * s + khalf];
            a.y = sW[(o0 + mrow) * SWS + 4 * s + khalf + 1];
            // B: 4x16 fragment of x[4s+k][n0+n]
            bb.x = sX[(4 * s + khalf) * SXS + n0 + mrow];
            bb.y = sX[(4 * s + khalf + 1) * SXS + n0 + mrow];
            acc = __builtin_amdgcn_wmma_f32_16x16x4_f32(
                false, a, false, bb, (short)0, acc, false, false);
        }
        // C/D layout: VGPR r -> rows r (lanes 0-15) and r+8 (lanes 16-31)
        int chbase = o0 + ((lane >> 4) << 3);
        #pragma unroll
        for (int r = 0; r < 8; ++r) {
            int ch = chbase + r;
            float g = acc[r] * sS[ch] + sB[ch];
            float v = silu_f(g);
            vg += sG[ch] * v;
            vm += sM[ch] * v;
        }
    }
    // combine lane l (channels 0..7 of tile) with lane l+16 (channels 8..15)
    vg += __shfl_xor(vg, 16);
    vm += __shfl_xor(vm, 16);
    if (lane < 16) {
        size_t p = (size_t)b * NHW + base + n0 + lane;
        vgap[p] = vg;
        vgmp[p] = vm;
    }
    // xs: spatial-attention 64->1 projection
    if (tid < 128) {
        float cs = convAs[0], cb = convAb[0];
        float s = 0.0f;
        #pragma unroll
        for (int c = 0; c < NC; ++c) s += sCA[c] * sX[c * SXS + tid];
        xs[(size_t)b * NHW + base + tid] = silu_f(s * cs + cb);
    }
}

// ---------------------------------------------------------------------------
// K4a/K4b: morphology. erode = min over window (OOB -> +1e4);
// dilate_sub: out = orig - max over window of eroded (OOB -> -1e4).
// Matches reference geodesic-pad semantics exactly.
// ---------------------------------------------------------------------------
__global__ __launch_bounds__(256) void k_erode(const float* __restrict__ in,
                                               float* __restrict__ out,
                                               int rh, int rw) {
    int idx = blockIdx.x * 256 + threadIdx.x;       // over B*HW
    int b = idx >> 16, n = idx & 65535;
    int yy = n >> 8, xx = n & 255;
    const float* p = in + ((size_t)b << 16);
    float m = BIG;
    for (int dy = -rh; dy <= rh; ++dy)
        for (int dx = -rw; dx <= rw; ++dx) {
            int Y = yy + dy, X = xx + dx;
            float v = (Y >= 0 && Y < NH && X >= 0 && X < NW) ? p[Y * NW + X] : BIG;
            m = fminf(m, v);
        }
    out[idx] = m;
}

__global__ __launch_bounds__(256) void k_dilate_sub(const float* __restrict__ er,
                                                    const float* __restrict__ orig,
                                                    float* __restrict__ out,
                                                    int rh, int rw) {
    int idx = blockIdx.x * 256 + threadIdx.x;
    int b = idx >> 16, n = idx & 65535;
    int yy = n >> 8, xx = n & 255;
    const float* p = er + ((size_t)b << 16);
    float m = -BIG;
    for (int dy = -rh; dy <= rh; ++dy)
        for (int dx = -rw; dx <= rw; ++dx) {
            int Y = yy + dy, X = xx + dx;
            float v = (Y >= 0 && Y < NH && X >= 0 && X < NW) ? p[Y * NW + X] : -BIG;
            m = fmaxf(m, v);
        }
    out[idx] = orig[idx] - m;
}

// ---------------------------------------------------------------------------
// K5: spatial softmax over HW per batch. 1 block (1024 threads) per b.
// ---------------------------------------------------------------------------
__global__ __launch_bounds__(1024) void k_softmax_hw(const float* __restrict__ t,
                                                     float* __restrict__ sp) {
    __shared__ float red[1024];
    int b = blockIdx.x, tid = threadIdx.x;
    const float4* p4 = (const float4*)(t + ((size_t)b << 16));
    float mx = -3.4e38f;
    for (int i = tid; i < 16384; i += 1024) {
        float4 v = p4[i];
        mx = fmaxf(mx, fmaxf(fmaxf(v.x, v.y), fmaxf(v.z, v.w)));
    }
    red[tid] = mx; __syncthreads();
    for (int s = 512; s > 0; s >>= 1) { if (tid < s) red[tid] = fmaxf(red[tid], red[tid + s]); __syncthreads(); }
    mx = red[0]; __syncthreads();
    float sum = 0.0f;
    for (int i = tid; i < 16384; i += 1024) {
        float4 v = p4[i];
        sum += __expf(v.x - mx) + __expf(v.y - mx) + __expf(v.z - mx) + __expf(v.w - mx);
    }
    red[tid] = sum; __syncthreads();
    for (int s = 512; s > 0; s >>= 1) { if (tid < s) red[tid] += red[tid + s]; __syncthreads(); }
    float inv = 1.0f / red[0];
    float4* o4 = (float4*)(sp + ((size_t)b << 16));
    for (int i = tid; i < 16384; i += 1024) {
        float4 v = p4[i], r;
        r.x = __expf(v.x - mx) * inv; r.y = __expf(v.y - mx) * inv;
        r.z = __expf(v.z - mx) * inv; r.w = __expf(v.w - mx) * inv;
        o4[i] = r;
    }
}

// ---------------------------------------------------------------------------
// K6: GEMM pass 2 (WMMA). Recompute v per tile, contract with sp over pixels:
//   A2[b,c] += sum_n v[c,n] * sp[n]   (LDS partials + global f32 atomics)
// ---------------------------------------------------------------------------
__global__ __launch_bounds__(256) void k_gemm_pass2(
    const float* __restrict__ x,
    const float* __restrict__ convw, const float* __restrict__ convs,
    const float* __restrict__ convb,
    const float* __restrict__ sp, float* __restrict__ A2) {
    __shared__ float sW[NC * SWS];
    __shared__ float sX[NC * SXS];
    __shared__ float sS[NC], sB[NC], sSP[128], sA2[NC];

    int tid = threadIdx.x;
    int b = blockIdx.x >> 9;
    int chunk = blockIdx.x & 511;
    int base = chunk * 128;
    const float* xb = x + ((size_t)b * NC) * NHW;

    if (tid < NC) { sS[tid] = convs[tid]; sB[tid] = convb[tid]; sA2[tid] = 0.0f; }
    if (tid < 128) sSP[tid] = sp[(size_t)b * NHW + base + tid];
    stage_tiles(convw, xb, base, sW, sX, tid);
    __syncthreads();

    int lane = tid & 31;
    int wv = tid >> 5;
    int n0 = wv * 16;
    int mrow = lane & 15;
    int khalf = (lane >> 4) << 1;

    for (int t = 0; t < 4; ++t) {
        int o0 = t * 16;
        v8f acc = {};
        #pragma unroll
        for (int s = 0; s < 16; ++s) {
            v2f a, bb;
            a.x = sW[(o0 + mrow) * SWS + 4 * s + khalf];
            a.y = sW[(o0 + mrow) * SWS + 4 * s + khalf + 1];
            bb.x = sX[(4 * s + khalf) * SXS + n0 + mrow];
            bb.y = sX[(4 * s + khalf + 1) * SXS + n0 + mrow];
            acc = __builtin_amdgcn_wmma_f32_16x16x4_f32(
                false, a, false, bb, (short)0, acc, false, false);
        }
        int chbase = o0 + ((lane >> 4) << 3);
        #pragma unroll
        for (int r = 0; r < 8; ++r) {
            int ch = chbase + r;
            float g = acc[r] * sS[ch] + sB[ch];
            float ppart = silu_f(g) * sSP[n0 + mrow];
            // sum over the 16 pixels of this row (stays within a 16-lane half)
            ppart += __shfl_xor(ppart, 1);
            ppart += __shfl_xor(ppart, 2);
            ppart += __shfl_xor(ppart, 4);
            ppart += __shfl_xor(ppart, 8);
            if ((lane & 15) == 0) atomicAdd(&sA2[ch], ppart);
        }
    }
    __syncthreads();
    if (tid < NC) atomicAdd(&A2[b * NC + tid], sA2[tid]);
}

// ---------------------------------------------------------------------------
// K7: m[b,o] = silu(sum_c A2[b,c] * m1_w[o,c]). 512 outputs, one block.
// ---------------------------------------------------------------------------
__global__ void k_m1(const float* __restrict__ A2, const float* __restrict__ m1w,
                     float* __restrict__ m) {
    int i = threadIdx.x;            // 512 threads
    int b = i >> 6, o = i & 63;
    float s = 0.0f;
    #pragma unroll
    for (int c = 0; c < NC; ++c) s += A2[b * NC + c] * m1w[o * NC + c];
    m[i] = silu_f(s);
}

// ---------------------------------------------------------------------------
// K8: final (float4): A_weight = sigmoid(silu((w0*vgap + w1*vgmp)*s1 + b1));
//                     out[b,c,n] = m[b,c]*A_weight[b,n] + x[b,c,n]
// ---------------------------------------------------------------------------
__global__ __launch_bounds__(256) void k_final(
    const float* __restrict__ x, const float* __restrict__ vgap,
    const float* __restrict__ vgmp,
    const float* __restrict__ c1w, const float* __restrict__ c1s,
    const float* __restrict__ c1b,
    const float* __restrict__ m, float* __restrict__ out) {
    int idx4 = blockIdx.x * 256 + threadIdx.x;   // over B*HW/4
    int b = idx4 >> 14;                          // 16384 float4 per batch
    int n4 = idx4 & 16383;
    __shared__ float sm[NC];
    if (threadIdx.x < NC) sm[threadIdx.x] = m[b * NC + threadIdx.x];
    __syncthreads();
    float w0 = c1w[0], w1 = c1w[1], s1 = c1s[0], b1 = c1b[0];
    float4 vg = ((const float4*)vgap)[idx4];
    float4 vm = ((const float4*)vgmp)[idx4];
    float gv[4] = { vg.x, vg.y, vg.z, vg.w };
    float mv[4] = { vm.x, vm.y, vm.z, vm.w };
    float aw[4];
    #pragma unroll
    for (int j = 0; j < 4; ++j) {
        float g = (w0 * gv[j] + w1 * mv[j]) * s1 + b1;
        aw[j] = 1.0f / (1.0f + __expf(-silu_f(g)));
    }
    const float4* x4 = (const float4*)x;
    float4* o4 = (float4*)out;
    size_t bo = ((size_t)b << 20) + (size_t)n4;  // b * 64 * 16384 + n4
    #pragma unroll 4
    for (int c = 0; c < NC; ++c) {
        size_t p = bo + ((size_t)c << 14);
        float4 xv = x4[p];
        float mc = sm[c];
        float4 r;
        r.x = mc * aw[0] + xv.x; r.y = mc * aw[1] + xv.y;
        r.z = mc * aw[2] + xv.z; r.w = mc * aw[3] + xv.w;
        o4[p] = r;
    }
}

// ---------------------------------------------------------------------------
extern "C" void kernel_launch(void* const* d_in, const int* in_sizes, int n_in,
                              void* d_out, int out_size, void* d_ws, size_t ws_size,
                              hipStream_t stream) {
    const float* x      = (const float*)d_in[0];   // [8,64,256,256]
    const float* convw  = (const float*)d_in[1];   // [64,64]
    const float* convs  = (const float*)d_in[2];   // [64]
    const float* convb  = (const float*)d_in[3];   // [64]
    const float* c1w    = (const float*)d_in[4];   // [1,2]
    const float* c1s    = (const float*)d_in[5];   // [1]
    const float* c1b    = (const float*)d_in[6];   // [1]
    const float* cAw    = (const float*)d_in[7];   // [1,64]
    const float* cAs    = (const float*)d_in[8];   // [1]
    const float* cAb    = (const float*)d_in[9];   // [1]
    const float* m1w    = (const float*)d_in[10];  // [64,64]
    float* out = (float*)d_out;

    // workspace layout (floats); total ~12.6 MB
    float* w = (float*)d_ws;
    float* gapraw = w;                  // 512
    float* gmpraw = w + 512;            // 512
    float* gap    = w + 1024;           // 512
    float* gmp    = w + 1536;           // 512
    float* A2     = w + 2048;           // 512
    float* mbuf   = w + 2560;           // 512
    float* xs     = w + 4096;           // 524288
    float* ebuf   = xs + NB * NHW;      // 524288
    float* tbuf   = ebuf + NB * NHW;    // 524288
    float* vgap   = tbuf + NB * NHW;    // 524288
    float* vgmp   = vgap + NB * NHW;    // 524288
    float* sp     = vgmp + NB * NHW;    // 524288

    const int PIXBLK = NB * NHW / 256;  // 2048 blocks for per-pixel kernels

    k_zero<<<2, 256, 0, stream>>>(A2, NB * NC);
    k_stats<<<NB * NC, 256, 0, stream>>>(x, gapraw, gmpraw);
    k_softmax_c<<<NB, NC, 0, stream>>>(gapraw, gmpraw, gap, gmp);
    k_gemm_pass1<<<NB * 512, 256, 0, stream>>>(x, convw, convs, convb,
                                               cAw, cAs, cAb, gap, gmp,
                                               vgap, vgmp, xs);
    // tophat (5,1)
    k_erode<<<PIXBLK, 256, 0, stream>>>(xs, ebuf, 2, 0);
    k_dilate_sub<<<PIXBLK, 256, 0, stream>>>(ebuf, xs, tbuf, 2, 0);
    // tophat (3,3)
    k_erode<<<PIXBLK, 256, 0, stream>>>(tbuf, ebuf, 1, 1);
    k_dilate_sub<<<PIXBLK, 256, 0, stream>>>(ebuf, tbuf, xs, 1, 1);
    // tophat (1,5)
    k_erode<<<PIXBLK, 256, 0, stream>>>(xs, ebuf, 0, 2);
    k_dilate_sub<<<PIXBLK, 256, 0, stream>>>(ebuf, xs, tbuf, 0, 2);

    k_softmax_hw<<<NB, 1024, 0, stream>>>(tbuf, sp);
    k_gemm_pass2<<<NB * 512, 256, 0, stream>>>(x, convw, convs, convb, sp, A2);
    k_m1<<<1, 512, 0, stream>>>(A2, m1w, mbuf);
    k_final<<<PIXBLK / 4, 256, 0, stream>>>(x, vgap, vgmp, c1w, c1s, c1b, mbuf, out);
}